// GraphAttentionLayer_6949257085383
// MI455X (gfx1250) — compile-verified
//
#include <hip/hip_runtime.h>

typedef __attribute__((ext_vector_type(2))) float v2f;
typedef __attribute__((ext_vector_type(4))) float v4f;
typedef __attribute__((ext_vector_type(8))) float v8f;

// B=16 V=20 MF=32 MT=8 NP=4 K=3 R=12 D=128 NOUT=64
// One wave per (b,v,t, half-of-MF): 16 rows x D=128. grid = 16*20*8*2 = 5120.
//
// Phase A (WMMA): per row, rels[(p,k), d] = idx[12x12] @ diag[12x128] as
//   3 K-steps x 8 N-tiles of v_wmma_f32_16x16x4_f32. The 12 (p,k) rows are
//   REMAPPED in LDS to {0-2:p0, 3-5:p1, 6-7:zero, 8-10:p2, 11-13:p3, 14-15:zero}
//   so each path's K-product lives entirely in one wave-half of the D layout
//   (no cross-half bpermutes). Softmax over NP=4 (s.w + bias shift cancels).
// Phase B (WMMA): z[16x128] @ out_w[128x64] -> bias+relu -> row-reduce.
// Staging uses GLOBAL_LOAD_ASYNC_TO_LDS_B128 (ASYNCcnt) to bypass VGPRs.

__device__ __forceinline__ void async_ld16(unsigned lds_off, unsigned g_off,
                                           const float* base) {
  asm volatile("global_load_async_to_lds_b128 %0, %1, %2"
               :: "v"(lds_off), "v"(g_off), "s"(base)
               : "memory");
}

__launch_bounds__(32)
__global__ void gat_main(const float* __restrict__ se,      // [B,V,MT,MF,D]
                         const float* __restrict__ ridx,    // [B,V,MF,MT,NP,K,R]
                         const float* __restrict__ Wrel,    // [R,128,128]
                         const float* __restrict__ attn_w,  // [192]
                         const float* __restrict__ out_w,   // [128,64]
                         const float* __restrict__ out_b,   // [64]
                         float* __restrict__ partial)       // [5120,64]
{
  __shared__ float idx_t[16 * 192];   // per-row 16 M-rows x 12 r (remapped, padded)
  __shared__ float se_t[16 * 128];    // source_embed tile
  __shared__ float z_t[16 * 132];     // z rows for phase B (stride 132)

  const int lane = threadIdx.x;
  const int bid  = blockIdx.x;
  const int half = bid & 1;
  const int t    = (bid >> 1) & 7;
  const int bv   = bid >> 4;
  const int f0   = half * 16;
  const int col  = lane & 15;
  const int hi   = lane >> 4;         // 0: paths 0,1  | 1: paths 2,3

  // ---- zero pad rows m = 6,7,14,15 of each idx row-block (ds stores) ----
  {
    v4f z4 = {0.f, 0.f, 0.f, 0.f};
    for (int q = lane; q < 16 * 12; q += 32) {     // 4 pad rows x 3 float4 per rr
      const int rr = q / 12, w = q % 12;
      const int pm = w / 3, j = w % 3;
      const int m  = (pm < 2) ? (6 + pm) : (12 + pm);   // 6,7,14,15
      reinterpret_cast<v4f*>(&idx_t[rr * 192 + m * 12])[j] = z4;
    }
  }

  // ---- async stage rel_index tile with row remap: src row m -> lds row m+2*(m>=6) ----
  for (int q = lane; q < 16 * 36; q += 32) {       // 16 rr x 12 rows x 3 float4
    const int rr = q / 36, rem = q % 36;
    const int m = rem / 3, j = rem % 3;
    const int mp = m + ((m >= 6) ? 2 : 0);
    const unsigned lds_off = (unsigned)(size_t)&idx_t[rr * 192 + mp * 12 + j * 4];
    const unsigned g_off =
        (unsigned)((((bv * 32 + f0 + rr) * 8 + t) * 144 + m * 12 + j * 4) * 4);
    async_ld16(lds_off, g_off, ridx);
  }
  // ---- async stage source_embed tile (16 x 128 contiguous) ----
  {
    const unsigned g0 = (unsigned)((((bv * 8 + t) * 32 + f0) * 128) * 4);
    for (int q = lane; q < 16 * 32; q += 32) {
      const unsigned lds_off = (unsigned)(size_t)&se_t[q * 4];
      async_ld16(lds_off, g0 + (unsigned)(q * 16), se);
    }
  }

  // ---- overlap: gather B fragments (diag of W_rel) while async copies run ----
  v2f bfr[3][8];
  #pragma unroll
  for (int ks = 0; ks < 3; ++ks) {
    const int kk = ks * 4 + 2 * hi;
    #pragma unroll
    for (int nt = 0; nt < 8; ++nt) {
      const int d = nt * 16 + col;
      v2f b;
      b.x = Wrel[(size_t)kk * 16384 + (size_t)d * 129];        // W_rel[r][d][d]
      b.y = Wrel[(size_t)(kk + 1) * 16384 + (size_t)d * 129];
      bfr[ks][nt] = b;
    }
  }
  float awf[8];
  #pragma unroll
  for (int nt = 0; nt < 8; ++nt) awf[nt] = attn_w[64 + nt * 16 + col];

  asm volatile("s_wait_asynccnt 0x0" ::: "memory");   // async LDS writes done
  __syncthreads();                                     // pad ds stores done

  // ---- Phase A ----
  for (int rr = 0; rr < 16; ++rr) {
    v2f af[3];
    #pragma unroll
    for (int ks = 0; ks < 3; ++ks)
      af[ks] = *reinterpret_cast<const v2f*>(&idx_t[rr * 192 + col * 12 + ks * 4 + 2 * hi]);

    float lg0 = 0.f, lg1 = 0.f;
    float zjA[8], zjB[8];
    #pragma unroll
    for (int nt = 0; nt < 8; ++nt) {
      v8f acc = {};
      acc = __builtin_amdgcn_wmma_f32_16x16x4_f32(false, af[0], false, bfr[0][nt], (short)0, acc, false, false);
      acc = __builtin_amdgcn_wmma_f32_16x16x4_f32(false, af[1], false, bfr[1][nt], (short)0, acc, false, false);
      acc = __builtin_amdgcn_wmma_f32_16x16x4_f32(false, af[2], false, bfr[2][nt], (short)0, acc, false, false);
      // D layout: VGPR v -> row m = v (low half) / v+8 (high half); N=col = d column.
      // Remap put each path's 3 rows in v0-2 / v3-5 of its own half.
      const float wdA = acc[0] * acc[1] * acc[2];   // p0 (low) / p2 (high)
      const float wdB = acc[3] * acc[4] * acc[5];   // p1 (low) / p3 (high)
      const float sev = se_t[rr * 128 + nt * 16 + col];
      const float za = wdA * sev;
      const float zb = wdB * sev;
      zjA[nt] = za;
      zjB[nt] = zb;
      lg0 += za * awf[nt];
      lg1 += zb * awf[nt];
    }
    // reduce logits over the 128 d-columns (within each 16-lane half)
    lg0 += __shfl_xor(lg0, 8, 32); lg0 += __shfl_xor(lg0, 4, 32);
    lg0 += __shfl_xor(lg0, 2, 32); lg0 += __shfl_xor(lg0, 1, 32);
    lg1 += __shfl_xor(lg1, 8, 32); lg1 += __shfl_xor(lg1, 4, 32);
    lg1 += __shfl_xor(lg1, 2, 32); lg1 += __shfl_xor(lg1, 1, 32);
    const float og0 = __shfl_xor(lg0, 16, 32);   // other half's two path logits
    const float og1 = __shfl_xor(lg1, 16, 32);
    // softmax over 4 paths (uniform s.w+b shift cancels)
    const float mx  = fmaxf(fmaxf(lg0, lg1), fmaxf(og0, og1));
    const float e0  = __expf(lg0 - mx);
    const float e1  = __expf(lg1 - mx);
    const float e2  = __expf(og0 - mx);
    const float e3  = __expf(og1 - mx);
    const float inv = 1.f / (e0 + e1 + e2 + e3);
    const float w0  = e0 * inv, w1 = e1 * inv;
    #pragma unroll
    for (int nt = 0; nt < 8; ++nt) {
      float zc = zjA[nt] * w0 + zjB[nt] * w1;      // my half's two paths
      zc += __shfl_xor(zc, 16, 32);                // + other half's two paths
      if (lane < 16) z_t[rr * 132 + nt * 16 + col] = zc;
    }
  }

  __syncthreads();

  // ---- Phase B: z[16x128] @ out_w[128x64] via v_wmma_f32_16x16x4_f32 ----
  v8f acc0 = {}, acc1 = {}, acc2 = {}, acc3 = {};
  const int khi = hi << 1;   // 0 or 2

  for (int k0 = 0; k0 < 128; k0 += 4) {
    const int kk = k0 + khi;
    const v2f a2 = *reinterpret_cast<const v2f*>(&z_t[col * 132 + kk]);
    {
      v2f b2; b2.x = out_w[kk * 64 + col];       b2.y = out_w[(kk + 1) * 64 + col];
      acc0 = __builtin_amdgcn_wmma_f32_16x16x4_f32(false, a2, false, b2, (short)0, acc0, false, false);
    }
    {
      v2f b2; b2.x = out_w[kk * 64 + 16 + col];  b2.y = out_w[(kk + 1) * 64 + 16 + col];
      acc1 = __builtin_amdgcn_wmma_f32_16x16x4_f32(false, a2, false, b2, (short)0, acc1, false, false);
    }
    {
      v2f b2; b2.x = out_w[kk * 64 + 32 + col];  b2.y = out_w[(kk + 1) * 64 + 32 + col];
      acc2 = __builtin_amdgcn_wmma_f32_16x16x4_f32(false, a2, false, b2, (short)0, acc2, false, false);
    }
    {
      v2f b2; b2.x = out_w[kk * 64 + 48 + col];  b2.y = out_w[(kk + 1) * 64 + 48 + col];
      acc3 = __builtin_amdgcn_wmma_f32_16x16x4_f32(false, a2, false, b2, (short)0, acc3, false, false);
    }
  }

  // ---- epilogue: bias + relu, reduce over 16 rows, write partial ----
  v8f accs[4] = {acc0, acc1, acc2, acc3};
  #pragma unroll
  for (int nt = 0; nt < 4; ++nt) {
    const int n = nt * 16 + col;
    const float bias = out_b[n];
    float local = 0.f;
    #pragma unroll
    for (int v = 0; v < 8; ++v) local += fmaxf(accs[nt][v] + bias, 0.f);
    local += __shfl_xor(local, 16, 32);   // combine M=0..7 with M=8..15
    if (lane < 16) partial[(size_t)bid * 64 + n] = local;
  }
}

// Sum 16 tile-partials (MT=8 x 2 halves) per (b,v) -> d_out[b,v,64]
__global__ void gat_reduce(const float* __restrict__ partial, float* __restrict__ out) {
  const int bv = blockIdx.x;
  const int n  = threadIdx.x;
  float sum = 0.f;
  #pragma unroll
  for (int i = 0; i < 16; ++i)
    sum += partial[((size_t)bv * 16 + i) * 64 + n];
  out[bv * 64 + n] = sum;
}

extern "C" void kernel_launch(void* const* d_in, const int* in_sizes, int n_in,
                              void* d_out, int out_size, void* d_ws, size_t ws_size,
                              hipStream_t stream) {
  const float* se     = (const float*)d_in[0];  // source_embed
  const float* ridx   = (const float*)d_in[1];  // rel_index
  // d_in[2] = s          : unused (constant across softmax axis -> cancels)
  const float* Wrel   = (const float*)d_in[3];  // W_rel
  const float* attn_w = (const float*)d_in[4];
  // d_in[5] = attn_b     : unused (cancels in softmax)
  const float* out_w  = (const float*)d_in[6];
  const float* out_b  = (const float*)d_in[7];

  float* partial = (float*)d_ws;                 // 5120*64*4 = 1.31 MB

  gat_main<<<5120, 32, 0, stream>>>(se, ridx, Wrel, attn_w, out_w, out_b, partial);
  gat_reduce<<<320, 64, 0, stream>>>(partial, (float*)d_out);
}